// TopicRouter_68573447848334
// MI455X (gfx1250) — compile-verified
//
#include <hip/hip_runtime.h>

typedef float v2f __attribute__((ext_vector_type(2)));
typedef float v8f __attribute__((ext_vector_type(8)));

#define D_DIM 768
#define E_DIM 8
#define W_STRIDE 772            // 768 + 4 pad; 772 % 64 == 4 -> conflict-free b64 B reads
#define WAVES_PER_BLOCK 8
#define ROWS_PER_WAVE 16
#define BLOCK_THREADS 256

__global__ __launch_bounds__(BLOCK_THREADS) void TopicRouter_gfx1250_kernel(
    const float* __restrict__ h,
    const float* __restrict__ gate_w,
    const float* __restrict__ gate_b,
    int*   __restrict__ idx_out,
    float* __restrict__ w_out,
    float* __restrict__ logits_out)
{
    __shared__ float sW[E_DIM * W_STRIDE];
    __shared__ float sLogits[WAVES_PER_BLOCK][ROWS_PER_WAVE][E_DIM];

    const int tid  = threadIdx.x;
    const int wave = tid >> 5;
    const int lane = tid & 31;

    // ---- Stage gate_w[8][768] into LDS with padded row stride (uniform loop) ----
    #pragma unroll
    for (int i = tid; i < E_DIM * D_DIM; i += BLOCK_THREADS) {
        int e = i / D_DIM;
        int k = i - e * D_DIM;
        sW[e * W_STRIDE + k] = gate_w[i];
    }
    __syncthreads();

    // ---- WMMA tile setup (wave32: 16x16x4 f32) ----
    const int n     = lane & 15;          // logits column owned by this lane (valid if < 8)
    const int khalf = (lane >> 4) * 2;    // which K sub-pair this half-wave provides
    const int tile  = blockIdx.x * WAVES_PER_BLOCK + wave;
    const int rowBase = tile * ROWS_PER_WAVE;

    // C initialized with the bias -> bias add is free (column n is constant per lane)
    const float bias = (n < E_DIM) ? gate_b[n] : 0.0f;
    v8f acc = {bias, bias, bias, bias, bias, bias, bias, bias};

    const float* aPtr = h + (size_t)(rowBase + (lane & 15)) * D_DIM + khalf;
    const float* bPtr = &sW[(n & 7) * W_STRIDE + khalf];   // cols 8..15 duplicate 0..7 (never read)

    #pragma unroll 4
    for (int k0 = 0; k0 < D_DIM; k0 += 4) {
        v2f a = *(const v2f*)(aPtr + k0);   // global_load_b64, 8B aligned
        v2f b = *(const v2f*)(bPtr + k0);   // ds_load_b64, conflict-free
        acc = __builtin_amdgcn_wmma_f32_16x16x4_f32(
                  /*neg_a=*/false, a, /*neg_b=*/false, b,
                  /*c_mod=*/(short)0, acc, /*reuse_a=*/false, /*reuse_b=*/false);
    }

    // ---- Epilogue: acc[r] = logits[rowBase + (lane>>4)*8 + r][n] ----
    const int rHalf = (lane >> 4) * 8;
    if (n < E_DIM) {
        #pragma unroll
        for (int r = 0; r < 8; ++r) {
            const int row = rowBase + rHalf + r;
            logits_out[(size_t)row * E_DIM + n] = acc[r];
            sLogits[wave][rHalf + r][n] = acc[r];
        }
    }
    __syncthreads();

    // ---- Top-2 + softmax: lanes 0..15 each own one row of the tile ----
    if (lane < ROWS_PER_WAVE) {
        const float* s = sLogits[wave][lane];
        float best = s[0], second = -__builtin_inff();
        int bi = 0, si = 0;
        #pragma unroll
        for (int e = 1; e < E_DIM; ++e) {
            float v = s[e];
            if (v > best)        { second = best; si = bi; best = v; bi = e; }
            else if (v > second) { second = v; si = e; }
        }
        const float e1 = __expf(second - best);
        const float w0 = 1.0f / (1.0f + e1);
        const float w1 = e1 * w0;
        const int row = rowBase + lane;
        idx_out[row * 2 + 0] = bi;
        idx_out[row * 2 + 1] = si;
        w_out[row * 2 + 0] = w0;
        w_out[row * 2 + 1] = w1;
    }
}

extern "C" void kernel_launch(void* const* d_in, const int* in_sizes, int n_in,
                              void* d_out, int out_size, void* d_ws, size_t ws_size,
                              hipStream_t stream) {
    const float* h      = (const float*)d_in[0];
    const float* gate_w = (const float*)d_in[1];
    const float* gate_b = (const float*)d_in[2];

    const int B = in_sizes[0] / D_DIM;              // 32768

    // d_out layout: indices [B][2] int32 | weights [B][2] f32 | logits [B][8] f32
    int*   idx_out    = (int*)d_out;
    float* w_out      = (float*)d_out + (size_t)B * 2;
    float* logits_out = (float*)d_out + (size_t)B * 4;

    const int tiles  = B / ROWS_PER_WAVE;           // 2048
    const int blocks = tiles / WAVES_PER_BLOCK;     // 256

    TopicRouter_gfx1250_kernel<<<blocks, BLOCK_THREADS, 0, stream>>>(
        h, gate_w, gate_b, idx_out, w_out, logits_out);
}